// Aggregation_45878840656648
// MI455X (gfx1250) — compile-verified
//
#include <hip/hip_runtime.h>

#ifndef __has_builtin
#define __has_builtin(x) 0
#endif

namespace {
constexpr int N_ = 8, C_ = 64, H_ = 128, W_ = 128;
constexpr int CW_ = 8, G_ = 8, K_ = 9;
constexpr int TH_ = 16, TW_ = 16;
constexpr int HALO_H = TH_ + 2, HALO_W = TW_ + 2;
constexpr int LDSW = 20;                    // padded LDS row stride (floats) -> no bank conflicts
constexpr int HW_ = H_ * W_;
}

// CDNA5 async global->LDS copy (ASYNCcnt). VDST VGPR holds the LDS byte
// address (low 32 bits of a generic pointer to __shared__), VADDR is the
// 64-bit global address. Inline asm: portable across ROCm7.2 / amdgpu-toolchain.
__device__ __forceinline__ void async_load_f32_to_lds(const float* gp,
                                                      unsigned lds_byte_off) {
  asm volatile("global_load_async_to_lds_b32 %0, %1, off"
               :: "v"(lds_byte_off), "v"(gp)
               : "memory");
}

__device__ __forceinline__ void wait_async_zero() {
#if __has_builtin(__builtin_amdgcn_s_wait_asynccnt)
  __builtin_amdgcn_s_wait_asynccnt(0);
#else
  asm volatile("s_wait_asynccnt 0" ::: "memory");
#endif
}

__global__ __launch_bounds__(256)
void san_aggregate_kernel(const float* __restrict__ x,
                          const float* __restrict__ w,
                          float* __restrict__ out) {
  // 8 channels x (18x18) halo tile, padded rows: 8*18*20*4 = 11.25 KB of 320 KB
  __shared__ float xs[G_][HALO_H][LDSW];

  const int tx  = blockIdx.x;
  const int ty  = blockIdx.y;
  const int n   = blockIdx.z / CW_;
  const int cw  = blockIdx.z % CW_;
  const int tid = threadIdx.x;

  const int x0 = tx * TW_;
  const int y0 = ty * TH_;

  const float* xb = x + (size_t)(n * C_ + cw * G_) * HW_;

  // ---- stage the 8-channel halo tile into LDS via async DMA (2592 lanes) ----
  #pragma unroll 1
  for (int idx = tid; idx < G_ * HALO_H * HALO_W; idx += 256) {
    const int g  = idx / (HALO_H * HALO_W);
    const int r  = idx - g * (HALO_H * HALO_W);
    const int ly = r / HALO_W;
    const int lx = r - ly * HALO_W;
    const int gy = y0 + ly - 1;
    const int gx = x0 + lx - 1;
    float* lp = &xs[g][ly][lx];
    if ((unsigned)gy < (unsigned)H_ && (unsigned)gx < (unsigned)W_) {
      async_load_f32_to_lds(xb + (size_t)g * HW_ + gy * W_ + gx,
                            (unsigned)(unsigned long long)lp);
    } else {
      *lp = 0.0f;                      // zero padding at image borders
    }
  }

  // ---- per-pixel 9-tap dynamic weights: independent stream, overlaps DMA ----
  const int ly = tid >> 4;             // 16x16 pixel tile, one pixel per thread
  const int lx = tid & 15;
  const int gy = y0 + ly;
  const int gx = x0 + lx;
  const int pix = gy * W_ + gx;
  const float* wb = w + ((size_t)(n * CW_ + cw) * K_) * HW_ + pix;

  float wk[K_];
  #pragma unroll
  for (int k = 0; k < K_; ++k) wk[k] = wb[(size_t)k * HW_];

  wait_async_zero();                   // this wave's async loads landed in LDS
  __syncthreads();                     // all waves' tiles visible

  // ---- 8 group channels x 9 taps, x read 9x from LDS instead of HBM ----
  float* ob = out + (size_t)(n * C_ + cw * G_) * HW_ + pix;
  #pragma unroll
  for (int g = 0; g < G_; ++g) {
    float acc = 0.0f;
    #pragma unroll
    for (int i = 0; i < 3; ++i) {
      #pragma unroll
      for (int j = 0; j < 3; ++j) {
        acc = fmaf(xs[g][ly + i][lx + j], wk[3 * i + j], acc);
      }
    }
    ob[(size_t)g * HW_] = acc;
  }
}

extern "C" void kernel_launch(void* const* d_in, const int* in_sizes, int n_in,
                              void* d_out, int out_size, void* d_ws, size_t ws_size,
                              hipStream_t stream) {
  const float* x  = (const float*)d_in[0];   // [8, 64, 128, 128] fp32
  const float* w  = (const float*)d_in[1];   // [8, 8, 9, 16384]  fp32
  float* out      = (float*)d_out;           // [8, 64, 128, 128] fp32

  dim3 grid(W_ / TW_, H_ / TH_, N_ * CW_);   // (8, 8, 64) = 4096 blocks
  san_aggregate_kernel<<<grid, 256, 0, stream>>>(x, w, out);
}